// OccGridAccelBatchedDynamic_Getter_21242908246602
// MI455X (gfx1250) — compile-verified
//
#include <hip/hip_runtime.h>

#define NUM_FRAMES 32
#define RES 128
#define NPER 4          // points per thread
#define BLOCK 256       // 8 wave32s per block

// Nearest keyframe via faithful searchsorted (lower_bound) + tie-to-left,
// over LDS-resident sorted keyframes (n == 32, power of two).
__device__ __forceinline__ int nearest_kf(const float* __restrict__ skf, float q) {
    int idx = 0;
    // branchless lower_bound over 32 elements: count of elements < q
    #pragma unroll
    for (int step = 16; step > 0; step >>= 1) {
        if (skf[idx + step - 1] < q) idx += step;
    }
    // clip(idx, 1, n-1): search above yields idx in [0, 31]
    if (idx < 1) idx = 1;
    float left  = skf[idx - 1];
    float right = skf[idx];
    return (q - left <= right - q) ? (idx - 1) : idx;
}

__device__ __forceinline__ int quant_axis(float p) {
    // ((p/2 + 0.5) * RES) truncated toward zero, clipped to [0, RES-1]
    float v = (p * 0.5f + 0.5f) * (float)RES;
    int g = (int)v;                 // round-toward-zero, matches astype(int32)
    g = (g < 0) ? 0 : g;
    g = (g > RES - 1) ? (RES - 1) : g;
    return g;
}

__device__ __forceinline__ float gather_one(const float* __restrict__ occ,
                                            const float* __restrict__ skf,
                                            float px, float py, float pz,
                                            float t, int b) {
    int gx = quant_axis(px);
    int gy = quant_axis(py);
    int gz = quant_axis(pz);
    int f  = nearest_kf(skf, t);
    long long cell = ((long long)(b * NUM_FRAMES + f) << 21)
                   | ((long long)gx << 14) | ((long long)gy << 7) | (long long)gz;
    // Random gather over 1 GiB table: zero expected reuse -> non-temporal
    // (global_load_b32 th:TH_LOAD_NT on gfx1250), keep L2 for the streams.
    return __builtin_nontemporal_load(occ + cell);
}

__global__ __launch_bounds__(BLOCK)
void occ_grid_gather_kernel(const float* __restrict__ occ,
                            const float* __restrict__ pts,
                            const float* __restrict__ ts,
                            const float* __restrict__ kf,
                            const int*   __restrict__ bidx,
                            float*       __restrict__ out,
                            int n) {
    __shared__ float skf[NUM_FRAMES];
    if (threadIdx.x < NUM_FRAMES) skf[threadIdx.x] = kf[threadIdx.x];
    __syncthreads();

    int base = (blockIdx.x * BLOCK + threadIdx.x) * NPER;

    if (base + NPER <= n) {
        // Fully vectorized streaming loads (b128 each)
        float4 t4 = *(const float4*)(ts + base);
        int4   b4 = *(const int4*)(bidx + base);
        const float* pp = pts + 3 * (long long)base;
        float4 p0 = *(const float4*)(pp + 0);   // x0 y0 z0 x1
        float4 p1 = *(const float4*)(pp + 4);   // y1 z1 x2 y2
        float4 p2 = *(const float4*)(pp + 8);   // z2 x3 y3 z3

        float4 r;
        r.x = gather_one(occ, skf, p0.x, p0.y, p0.z, t4.x, b4.x);
        r.y = gather_one(occ, skf, p0.w, p1.x, p1.y, t4.y, b4.y);
        r.z = gather_one(occ, skf, p1.z, p1.w, p2.x, t4.z, b4.z);
        r.w = gather_one(occ, skf, p2.y, p2.z, p2.w, t4.w, b4.w);

        *(float4*)(out + base) = r;
    } else {
        // Tail (not hit for NUM_PTS = 4M, but keep it correct/deterministic)
        for (int k = 0; k < NPER; ++k) {
            int i = base + k;
            if (i < n) {
                float px = pts[3 * (long long)i + 0];
                float py = pts[3 * (long long)i + 1];
                float pz = pts[3 * (long long)i + 2];
                out[i] = gather_one(occ, skf, px, py, pz, ts[i], bidx[i]);
            }
        }
    }
}

extern "C" void kernel_launch(void* const* d_in, const int* in_sizes, int n_in,
                              void* d_out, int out_size, void* d_ws, size_t ws_size,
                              hipStream_t stream) {
    const float* occ  = (const float*)d_in[0];  // (B*F, 128,128,128) f32
    const float* pts  = (const float*)d_in[1];  // (N, 3) f32
    const float* ts   = (const float*)d_in[2];  // (N,)  f32
    const float* kf   = (const float*)d_in[3];  // (32,) f32
    const int*   bidx = (const int*)d_in[4];    // (N,)  int32
    float*       out  = (float*)d_out;          // (N,)  f32

    int n = in_sizes[2];                        // NUM_PTS
    int threads_needed = (n + NPER - 1) / NPER;
    int blocks = (threads_needed + BLOCK - 1) / BLOCK;

    occ_grid_gather_kernel<<<blocks, BLOCK, 0, stream>>>(occ, pts, ts, kf, bidx, out, n);
}